// GCN_62345745268793
// MI455X (gfx1250) — compile-verified
//
#include <hip/hip_runtime.h>
#include <stdint.h>

typedef __attribute__((ext_vector_type(2))) float v2f;
typedef __attribute__((ext_vector_type(8))) float v8f;

#define NFEAT  256
#define NHID   64
#define NCLASS 40

// ---------------------------------------------------------------- zero fill
__global__ void k_zero(float* __restrict__ p, int n) {
    int i = blockIdx.x * blockDim.x + threadIdx.x;
    int stride = gridDim.x * blockDim.x;
    for (; i < n; i += stride) p[i] = 0.0f;
}

// ---------------------------------------------------------------- B repack
// Pack B[K x N] into WMMA-fragment order, zero-padded to NP = NT*16 columns:
//   Bp[ ((k0/4)*2 + h) * NP + c ] = ( B[k0+2h, c], B[k0+2h+1, c] )  as v2f
// so the GEMM inner loop reads one contiguous b64 per fragment.
template <int K, int N, int NT>
__global__ void k_packB(const float* __restrict__ B, float* __restrict__ Bp) {
    constexpr int NP = NT * 16;
    int idx = blockIdx.x * blockDim.x + threadIdx.x;
    if (idx >= (K / 4) * 2 * NP) return;
    int c  = idx % NP;
    int rh = idx / NP;          // (k0/4)*2 + h
    int h  = rh & 1;
    int k0 = (rh >> 1) * 4;
    float x0 = (c < N) ? B[(k0 + 2 * h) * N + c]     : 0.0f;
    float x1 = (c < N) ? B[(k0 + 2 * h + 1) * N + c] : 0.0f;
    Bp[2 * idx]     = x0;
    Bp[2 * idx + 1] = x1;
}

// ---------------------------------------------------------------- WMMA GEMM
// C[M x N] = A[M x K] @ B[K x N] (B pre-packed), via V_WMMA_F32_16X16X4_F32.
// One wave per 16-row strip; NT 16-column accumulator tiles kept in VGPRs so
// the A fragment is loaded once per k-step and reused across all N tiles.
// Inner loop: 1 + NT global_load_b64 and NT wmma — no branches, no selects.
// Requires M % 16 == 0 (100000 = 6250*16), K % 4 == 0.
//
// Fragment layouts (CDNA5 ISA 7.12.2, wave32):
//   A (16x4):  lane l -> row m0+(l&15); VGPR v, half h=(l>=16) -> K = k0+v+2h
//   B (4x16):  VGPR v, half h -> K = k0+v+2h ; lane l -> col n0+(l&15)
//   C/D(16x16):VGPR v, half h -> row m0+v+8h ; lane l -> col n0+(l&15)
template <int K, int N, int NT>
__global__ void k_gemm_wmma(const float* __restrict__ A,
                            const float* __restrict__ Bp,
                            float* __restrict__ C, int M) {
    constexpr int NP = NT * 16;
    const int lane = threadIdx.x & 31;
    const int wave = threadIdx.x >> 5;
    const int half = lane >> 4;       // 0: lanes 0-15, 1: lanes 16-31
    const int l15  = lane & 15;
    const int m0   = (blockIdx.x * (blockDim.x >> 5) + wave) * 16;
    if (m0 >= M) return;              // wave-uniform -> EXEC all-1s inside

    v8f acc[NT] = {};

    const float* arow = A + (size_t)(m0 + l15) * K + 2 * half;
    const v2f*   bp   = (const v2f*)Bp + (size_t)half * NP + l15;

#pragma unroll 4
    for (int k0 = 0; k0 < K; k0 += 4) {
        v2f a = *(const v2f*)(arow + k0);                 // A[r, k0+2h ..+1]
        const v2f* brow = bp + (size_t)(k0 >> 2) * 2 * NP;
#pragma unroll
        for (int t = 0; t < NT; ++t) {
            v2f b = brow[t * 16];                         // packed B pair
            acc[t] = __builtin_amdgcn_wmma_f32_16x16x4_f32(
                false, a, false, b, (short)0, acc[t], false, false);
        }
    }

#pragma unroll
    for (int t = 0; t < NT; ++t) {
        int c = t * 16 + l15;
        if (c < N) {                                      // compile-time for N=64
#pragma unroll
            for (int v = 0; v < 8; ++v)
                C[(size_t)(m0 + v + 8 * half) * N + c] = acc[t][v];
        }
    }
}

// ---------------------------------------------------------------- SpMM scatter
// One wave (32 lanes) per edge: agg[dst,:] += w * support[src,:]
__global__ void k_spmm(const float* __restrict__ sup,
                       const int64_t* __restrict__ src,
                       const int64_t* __restrict__ dst,
                       const float* __restrict__ w,
                       float* __restrict__ agg, int E, int F) {
    int gw   = (blockIdx.x * blockDim.x + threadIdx.x) >> 5;
    int lane = threadIdx.x & 31;
    if (gw >= E) return;
    int64_t s = src[gw];
    int64_t d = dst[gw];
    float  wt = w[gw];
    const float* sp = sup + (size_t)s * F;
    float*       ap = agg + (size_t)d * F;
    for (int f = lane; f < F; f += 32)
        atomicAdd(ap + f, sp[f] * wt);       // global_atomic_add_f32
}

// ---------------------------------------------------------------- bias + ReLU (F == 64)
__global__ void k_bias_relu64(float* __restrict__ h, const float* __restrict__ b, int n) {
    int i = blockIdx.x * blockDim.x + threadIdx.x;
    int stride = gridDim.x * blockDim.x;
    for (; i < n; i += stride) {
        float v = h[i] + b[i & 63];
        h[i] = v > 0.0f ? v : 0.0f;
    }
}

// ---------------------------------------------------------------- bias + log_softmax (40 classes)
// One wave per row; lane handles class `lane` and (if lane<8) class `lane+32`.
__global__ void k_bias_logsoftmax(const float* __restrict__ agg,
                                  const float* __restrict__ b,
                                  float* __restrict__ out, int M) {
    int row  = (blockIdx.x * blockDim.x + threadIdx.x) >> 5;
    int lane = threadIdx.x & 31;
    if (row >= M) return;
    const float* a = agg + (size_t)row * NCLASS;

    bool  has1 = lane < (NCLASS - 32);
    float v0 = a[lane] + b[lane];
    float v1 = has1 ? (a[lane + 32] + b[lane + 32]) : -3.402823e38f;

    float m = fmaxf(v0, v1);
#pragma unroll
    for (int off = 16; off > 0; off >>= 1)
        m = fmaxf(m, __shfl_xor(m, off, 32));

    float s = expf(v0 - m) + (has1 ? expf(v1 - m) : 0.0f);
#pragma unroll
    for (int off = 16; off > 0; off >>= 1)
        s += __shfl_xor(s, off, 32);

    float ls = logf(s);
    float* o = out + (size_t)row * NCLASS;
    o[lane] = v0 - m - ls;
    if (has1) o[lane + 32] = v1 - m - ls;
}

// ---------------------------------------------------------------- launch
extern "C" void kernel_launch(void* const* d_in, const int* in_sizes, int n_in,
                              void* d_out, int out_size, void* d_ws, size_t ws_size,
                              hipStream_t stream) {
    const float*   x   = (const float*)d_in[0];
    const int64_t* ei  = (const int64_t*)d_in[1];
    const float*   ew  = (const float*)d_in[2];
    const float*   W1  = (const float*)d_in[3];
    const float*   b1  = (const float*)d_in[4];
    const float*   W2  = (const float*)d_in[5];
    const float*   b2  = (const float*)d_in[6];
    const float*   W3  = (const float*)d_in[7];
    const float*   b3  = (const float*)d_in[8];
    float*         out = (float*)d_out;

    const int M = in_sizes[0] / NFEAT;   // 100000 (multiple of 16)
    const int E = in_sizes[2];           // 1600000
    const int64_t* src = ei;             // edge_index[0]
    const int64_t* dst = ei + E;         // edge_index[1]

    char*  ws   = (char*)d_ws;
    size_t secF = (size_t)M * NHID * sizeof(float);   // 25.6 MB
    float* sup  = (float*)(ws);                       // support buffer
    float* agg1 = (float*)(ws + secF);                // layer-1 agg / h1 ; reused for layer-3 agg
    float* agg2 = (float*)(ws + 2 * secF);            // layer-2 agg / h2
    float* Bp   = (float*)(ws + 3 * secF);            // packed weights (<= 64 KB)

    const dim3 blk(256);                               // 8 wave32 per block
    const int gemm_blocks = (M / 16 + 7) / 8;          // one 16-row strip per wave
    const int spmm_blocks = (E + 7) / 8;               // one edge per wave
    const int ew_blocks   = 2048;
    const int nF64        = M * NHID;

    // pack-kernel element counts: (K/4)*2*NP
    const int p1 = (NFEAT / 4) * 2 * 64;   // 8192
    const int p2 = (NHID  / 4) * 2 * 64;   // 2048
    const int p3 = (NHID  / 4) * 2 * 48;   // 1536

    // ---- layer 1: h1 = relu(A @ (x @ W1) + b1)
    k_zero<<<ew_blocks, blk, 0, stream>>>(agg1, nF64);
    k_packB<NFEAT, NHID, 4><<<(p1 + 255) / 256, blk, 0, stream>>>(W1, Bp);
    k_gemm_wmma<NFEAT, NHID, 4><<<gemm_blocks, blk, 0, stream>>>(x, Bp, sup, M);
    k_spmm<<<spmm_blocks, blk, 0, stream>>>(sup, src, dst, ew, agg1, E, NHID);
    k_bias_relu64<<<ew_blocks, blk, 0, stream>>>(agg1, b1, nF64);

    // ---- layer 2: h2 = relu(A @ (h1 @ W2) + b2)
    k_zero<<<ew_blocks, blk, 0, stream>>>(agg2, nF64);
    k_packB<NHID, NHID, 4><<<(p2 + 255) / 256, blk, 0, stream>>>(W2, Bp);
    k_gemm_wmma<NHID, NHID, 4><<<gemm_blocks, blk, 0, stream>>>(agg1, Bp, sup, M);
    k_spmm<<<spmm_blocks, blk, 0, stream>>>(sup, src, dst, ew, agg2, E, NHID);
    k_bias_relu64<<<ew_blocks, blk, 0, stream>>>(agg2, b2, nF64);

    // ---- layer 3: out = log_softmax(A @ (h2 @ W3) + b3)
    float* agg3 = agg1;                                // h1 dead after layer-2 GEMM
    k_zero<<<ew_blocks, blk, 0, stream>>>(agg3, M * NCLASS);
    k_packB<NHID, NCLASS, 3><<<(p3 + 255) / 256, blk, 0, stream>>>(W3, Bp);
    k_gemm_wmma<NHID, NCLASS, 3><<<gemm_blocks, blk, 0, stream>>>(agg2, Bp, sup, M);
    k_spmm<<<spmm_blocks, blk, 0, stream>>>(sup, src, dst, ew, agg3, E, NCLASS);
    k_bias_logsoftmax<<<(M + 7) / 8, blk, 0, stream>>>(agg3, b3, out, M);
}